// SIM_9199819948195
// MI455X (gfx1250) — compile-verified
//
#include <hip/hip_runtime.h>
#include <math.h>

#define EPS 1e-8f
constexpr int B = 512, S = 4096, D = 64, K = 100;
constexpr int H4D = 256;   // 4*D
constexpr int NH = 80;     // hidden
constexpr int MPAD = 112;  // 7*16 padded rows
constexpr int SROW = 65;   // LDS row stride (bank-conflict free)

typedef float v2f __attribute__((ext_vector_type(2)));
typedef float v8f __attribute__((ext_vector_type(8)));

// ---------------------------------------------------------------------------
// Kernel 1: cosine similarity, one wave per (b,s) row. HBM-bound streaming.
// grid (8, B), block 256 (8 waves). Each block covers 512 s values.
// ---------------------------------------------------------------------------
__global__ void sim_kernel(const float* __restrict__ cand,
                           const float* __restrict__ ub,
                           float* __restrict__ sim) {
  const int b = blockIdx.y;
  const int wave = threadIdx.x >> 5;
  const int lane = threadIdx.x & 31;

  float2 c2 = ((const float2*)(cand + (size_t)b * D))[lane];
  float cn2 = c2.x * c2.x + c2.y * c2.y;
  for (int off = 16; off > 0; off >>= 1) cn2 += __shfl_xor(cn2, off, 32);
  const float cnorm = fmaxf(sqrtf(cn2), EPS);

  const int sBase = blockIdx.x * 512;
  for (int i = 0; i < 64; ++i) {
    const int s = sBase + i * 8 + wave;
    float2 u2 = ((const float2*)(ub + ((size_t)b * S + s) * D))[lane];
    float dot = u2.x * c2.x + u2.y * c2.y;
    float n2 = u2.x * u2.x + u2.y * u2.y;
    for (int off = 16; off > 0; off >>= 1) {
      dot += __shfl_xor(dot, off, 32);
      n2  += __shfl_xor(n2,  off, 32);
    }
    if (lane == 0)
      sim[(size_t)b * S + s] = dot / (cnorm * fmaxf(sqrtf(n2), EPS));
  }
}

// ---------------------------------------------------------------------------
// Kernel 2: deterministic radix top-k set selection. grid B, block 256.
// ---------------------------------------------------------------------------
__global__ void topk_kernel(const float* __restrict__ sim,
                            int* __restrict__ topk) {
  __shared__ unsigned keys[S];
  __shared__ unsigned hist[256];
  __shared__ unsigned gtc[256], eqc[256];
  __shared__ unsigned s_prefix, s_remaining, s_ngt;

  const int t = threadIdx.x;
  const int b = blockIdx.x;

  for (int i = t; i < S; i += 256) {
    unsigned u = __float_as_uint(sim[(size_t)b * S + i]);
    keys[i] = (u & 0x80000000u) ? ~u : (u | 0x80000000u);
  }
  if (t == 0) { s_prefix = 0u; s_remaining = (unsigned)K; }
  __syncthreads();

  // 4-pass MSB radix select of the k-th key
  for (int pass = 0; pass < 4; ++pass) {
    const int shift = 24 - 8 * pass;
    const unsigned hiMask = (pass == 0) ? 0u : (0xFFFFFFFFu << (shift + 8));
    hist[t] = 0u;
    __syncthreads();
    const unsigned pfx = s_prefix;
    for (int i = t; i < S; i += 256) {
      const unsigned key = keys[i];
      if ((key & hiMask) == pfx) atomicAdd(&hist[(key >> shift) & 0xFFu], 1u);
    }
    __syncthreads();
    if (t == 0) {
      unsigned cum = 0, rem = s_remaining;
      for (int d = 255; d >= 0; --d) {
        const unsigned c = hist[d];
        if (cum + c >= rem) {
          s_remaining = rem - cum;
          s_prefix = pfx | ((unsigned)d << shift);
          break;
        }
        cum += c;
      }
    }
    __syncthreads();
  }

  const unsigned T = s_prefix;       // exact k-th key
  const unsigned rem = s_remaining;  // number of ==T entries to keep

  // deterministic compaction: per-thread counts over contiguous ranges + scan
  const int lo = t * (S / 256), hi = lo + (S / 256);
  unsigned g = 0, e = 0;
  for (int i = lo; i < hi; ++i) {
    const unsigned key = keys[i];
    g += (key > T);
    e += (key == T);
  }
  gtc[t] = g; eqc[t] = e;
  __syncthreads();
  if (t == 0) {
    unsigned run = 0;
    for (int j = 0; j < 256; ++j) { unsigned c = gtc[j]; gtc[j] = run; run += c; }
    s_ngt = run;
    run = 0;
    for (int j = 0; j < 256; ++j) { unsigned c = eqc[j]; eqc[j] = run; run += c; }
  }
  __syncthreads();
  const unsigned nGt = s_ngt;
  unsigned gp = gtc[t], ep = eqc[t];
  for (int i = lo; i < hi; ++i) {
    const unsigned key = keys[i];
    if (key > T) {
      topk[(size_t)b * K + (gp++)] = i;
    } else if (key == T) {
      if (ep < rem) topk[(size_t)b * K + nGt + ep] = i;
      ++ep;
    }
  }
}

// ---------------------------------------------------------------------------
// Kernel 3: fold W1 into Wc (128x80): Wc[0:64]=W1b-W1c, Wc[64:128]=W1d.
// ---------------------------------------------------------------------------
__global__ void wc_kernel(const float* __restrict__ W1, float* __restrict__ wc) {
  const int id = blockIdx.x * 256 + threadIdx.x;
  if (id < 128 * NH) {
    const int d = id / NH, n = id % NH;
    float v;
    if (d < 64)
      v = W1[(64 + d) * NH + n] - W1[(128 + d) * NH + n];
    else
      v = W1[(192 + (d - 64)) * NH + n];
    wc[id] = v;
  }
}

// ---------------------------------------------------------------------------
// Kernel 4: DIN attention via V_WMMA_F32_16X16X4_F32. grid B, block 256.
// H = relu([s, c*s] @ Wc + bias_b), logit = H @ W2 + b2, softmax, pool.
// ---------------------------------------------------------------------------
__global__ void attn_kernel(const float* __restrict__ cand,
                            const float* __restrict__ ub,
                            const float* __restrict__ W1,
                            const float* __restrict__ b1,
                            const float* __restrict__ W2,
                            const float* __restrict__ b2,
                            const int* __restrict__ topk,
                            const float* __restrict__ wc,
                            float* __restrict__ out) {
  __shared__ float Ssm[MPAD][SROW];   // gathered subseq (padded rows zero)
  __shared__ float Hpart[5][MPAD];    // per-N-tile logit partials
  __shared__ float csm[D];
  __shared__ float bias[NH];
  __shared__ float w2sm[NH];
  __shared__ float logits[MPAD];
  __shared__ float wts[MPAD];
  __shared__ float red[2];

  const int t = threadIdx.x;
  const int b = blockIdx.x;

  if (t < D) csm[t] = cand[(size_t)b * D + t];
  if (t < NH) w2sm[t] = W2[t];

  // gather top-k behavior rows into LDS (rows 100..111 zero-padded)
  {
    const int jr = t >> 6;   // 0..3
    const int d = t & 63;
    for (int j0 = 0; j0 < MPAD; j0 += 4) {
      const int j = j0 + jr;
      float v = 0.0f;
      if (j < K) {
        const int idx = topk[(size_t)b * K + j];
        v = ub[((size_t)b * S + idx) * D + d];
      }
      Ssm[j][d] = v;
    }
  }
  __syncthreads();

  // per-batch bias: b1 + c @ (W1a + W1c)
  if (t < NH) {
    float acc = b1[t];
    for (int d = 0; d < D; ++d)
      acc += csm[d] * (W1[d * NH + t] + W1[(128 + d) * NH + t]);
    bias[t] = acc;
  }
  __syncthreads();

  // --- WMMA GEMM: (112 x 128) @ (128 x 80), tiles 16x16, K-step 4 ---
  const int wave = t >> 5, lane = t & 31;
  const int hh = lane >> 4, r = lane & 15;

  for (int tile = wave; tile < 35; tile += 8) {
    const int mt = tile / 5, nt = tile % 5;
    const int m0 = mt * 16, n0 = nt * 16;
    const int m = m0 + r;       // A row for this lane
    const int n = n0 + r;       // B/D column for this lane
    v8f acc = {};

    // K = 0..63 : A = s
    for (int k0 = 0; k0 < 64; k0 += 4) {
      const int ka = k0 + 2 * hh;
      v2f a;  a.x = Ssm[m][ka];          a.y = Ssm[m][ka + 1];
      v2f bb; bb.x = wc[ka * NH + n];    bb.y = wc[(ka + 1) * NH + n];
      acc = __builtin_amdgcn_wmma_f32_16x16x4_f32(
          false, a, false, bb, (short)0, acc, false, false);
    }
    // K = 64..127 : A = c * s
    for (int k0 = 0; k0 < 64; k0 += 4) {
      const int ka = k0 + 2 * hh;
      v2f a;  a.x = csm[ka] * Ssm[m][ka];
              a.y = csm[ka + 1] * Ssm[m][ka + 1];
      v2f bb; bb.x = wc[(64 + ka) * NH + n];
              bb.y = wc[(64 + ka + 1) * NH + n];
      acc = __builtin_amdgcn_wmma_f32_16x16x4_f32(
          false, a, false, bb, (short)0, acc, false, false);
    }

    // epilogue: relu + multiply by W2 column, reduce the 16 n-lanes per row
    const float w2v = w2sm[n];
    for (int v = 0; v < 8; ++v) {
      const int mm = m0 + v + 8 * hh;
      float pv = (mm < K) ? fmaxf(acc[v] + bias[n], 0.0f) * w2v : 0.0f;
      for (int off = 1; off < 16; off <<= 1) pv += __shfl_xor(pv, off, 32);
      if (r == 0) Hpart[nt][m0 + v + 8 * hh] = pv;
    }
  }
  __syncthreads();

  // deterministic logit reduction over the 5 N-tiles
  if (t < MPAD) {
    float lg = b2[0];
    for (int nt = 0; nt < 5; ++nt) lg += Hpart[nt][t];
    logits[t] = lg;
  }
  __syncthreads();

  // softmax over the first K logits
  if (t == 0) {
    float mx = -1e30f;
    for (int j = 0; j < K; ++j) mx = fmaxf(mx, logits[j]);
    red[0] = mx;
  }
  __syncthreads();
  if (t < K) wts[t] = expf(logits[t] - red[0]);
  __syncthreads();
  if (t == 0) {
    float sm = 0.0f;
    for (int j = 0; j < K; ++j) sm += wts[j];
    red[1] = sm;
  }
  __syncthreads();

  // weighted pooling
  if (t < D) {
    const float inv = 1.0f / red[1];
    float acc = 0.0f;
    for (int j = 0; j < K; ++j) acc += wts[j] * Ssm[j][t];
    out[(size_t)b * D + t] = acc * inv;
  }
}

// ---------------------------------------------------------------------------
extern "C" void kernel_launch(void* const* d_in, const int* in_sizes, int n_in,
                              void* d_out, int out_size, void* d_ws, size_t ws_size,
                              hipStream_t stream) {
  const float* cand = (const float*)d_in[0];
  const float* ub   = (const float*)d_in[1];
  const float* W1   = (const float*)d_in[2];
  const float* b1   = (const float*)d_in[3];
  const float* W2   = (const float*)d_in[4];
  const float* b2   = (const float*)d_in[5];
  float* out = (float*)d_out;

  // workspace layout
  float* ws_sim = (float*)d_ws;                       // B*S floats (8 MB)
  int*   ws_topk = (int*)(ws_sim + (size_t)B * S);    // B*K ints
  float* ws_wc = (float*)(ws_topk + (size_t)B * K);   // 128*NH floats

  sim_kernel<<<dim3(8, B), 256, 0, stream>>>(cand, ub, ws_sim);
  topk_kernel<<<B, 256, 0, stream>>>(ws_sim, ws_topk);
  wc_kernel<<<(128 * NH + 255) / 256, 256, 0, stream>>>(W1, ws_wc);
  attn_kernel<<<B, 256, 0, stream>>>(cand, ub, W1, b1, W2, b2,
                                     ws_topk, ws_wc, out);
}